// CostFn_49692771615402
// MI455X (gfx1250) — compile-verified
//
#include <hip/hip_runtime.h>
#include <math.h>

typedef __attribute__((ext_vector_type(2))) float v2f;
typedef __attribute__((ext_vector_type(8))) float v8f;

// UR-robot DH / mass constants (folded at compile time; ca/sa are exact 0/±1)
static constexpr float kA[6]    = {0.0f, -0.425f, -0.3922f, 0.0f, 0.0f, 0.0f};
static constexpr float kD[6]    = {0.1625f, 0.0f, 0.0f, 0.1333f, 0.0997f, 0.0996f};
static constexpr float kCA[6]   = {0.0f, 1.0f, 1.0f, 0.0f, 0.0f, 1.0f};
static constexpr float kSA[6]   = {1.0f, 0.0f, 0.0f, 1.0f, -1.0f, 0.0f};
static constexpr float kMASS[6] = {3.761f, 8.058f, 2.846f, 1.37f, 1.3f, 0.365f};

// Block = 256 threads = 4 batches of 64 items. Each thread does one item's full
// FK + Jacobian + Cholesky; each of the first 4 waves reduces one batch's 64
// costs with a single V_WMMA_F32_16X16X4_F32 (A==ones -> layout-independent sum).
__global__ __launch_bounds__(256) void CostFn_kernel(const float* __restrict__ x,
                                                     float* __restrict__ out,
                                                     int nBatch, int totalFloats) {
    const int tid = threadIdx.x;
    __shared__ __align__(16) float xs[256 * 26];
    __shared__ float cbuf[256];

    // ---- Stage this block's 256 rows (26 floats each) with coalesced b128 loads
    {
        const float4* __restrict__ src = reinterpret_cast<const float4*>(x) + (size_t)blockIdx.x * (256 * 26 / 4);
        float4* dst = reinterpret_cast<float4*>(xs);
        const int base4 = blockIdx.x * (256 * 26 / 4);
        #pragma unroll
        for (int k = tid; k < 256 * 26 / 4; k += 256) {
            if ((base4 + k) * 4 + 3 < totalFloats) dst[k] = src[k];
        }
    }
    __syncthreads();

    const int item   = blockIdx.x * 256 + tid;
    const bool active = item < nBatch * 64;
    float cost = 0.0f;

    if (active) {
        const float* __restrict__ row = xs + tid * 26;
        float q[6];
        #pragma unroll
        for (int i = 0; i < 6; ++i) q[i] = row[6 + i];
        const float hx = row[19], hy = row[20], hz = row[21];

        // ---- Forward kinematics: affine T = [R|p], T <- T * DH(q_i)
        float P[7][3], Z[6][3];
        P[0][0] = P[0][1] = P[0][2] = 0.0f;
        Z[0][0] = 0.0f; Z[0][1] = 0.0f; Z[0][2] = 1.0f;
        float r00 = 1, r01 = 0, r02 = 0, r10 = 0, r11 = 1, r12 = 0, r20 = 0, r21 = 0, r22 = 1;
        float px = 0, py = 0, pz = 0;
        #pragma unroll
        for (int i = 0; i < 6; ++i) {
            float st, ct;
            sincosf(q[i], &st, &ct);
            const float ca = kCA[i], sa = kSA[i], aa = kA[i], dd = kD[i];
            // DH rotation columns + translation
            const float a0x = ct,        a0y = st;                       // a0z = 0
            const float a1x = -st * ca,  a1y = ct * ca,  const_a1z = sa; (void)const_a1z;
            const float a1z = sa;
            const float a2x = st * sa,   a2y = -ct * sa, a2z = ca;
            const float tx = aa * ct, ty = aa * st, tz = dd;
            const float n00 = r00 * a0x + r01 * a0y;
            const float n10 = r10 * a0x + r11 * a0y;
            const float n20 = r20 * a0x + r21 * a0y;
            const float n01 = r00 * a1x + r01 * a1y + r02 * a1z;
            const float n11 = r10 * a1x + r11 * a1y + r12 * a1z;
            const float n21 = r20 * a1x + r21 * a1y + r22 * a1z;
            const float n02 = r00 * a2x + r01 * a2y + r02 * a2z;
            const float n12 = r10 * a2x + r11 * a2y + r12 * a2z;
            const float n22 = r20 * a2x + r21 * a2y + r22 * a2z;
            px += r00 * tx + r01 * ty + r02 * tz;
            py += r10 * tx + r11 * ty + r12 * tz;
            pz += r20 * tx + r21 * ty + r22 * tz;
            r00 = n00; r01 = n01; r02 = n02;
            r10 = n10; r11 = n11; r12 = n12;
            r20 = n20; r21 = n21; r22 = n22;
            P[i + 1][0] = px; P[i + 1][1] = py; P[i + 1][2] = pz;
            if (i < 5) { Z[i + 1][0] = r02; Z[i + 1][1] = r12; Z[i + 1][2] = r22; }
        }

        // ---- Jacobian rows + Gram matrix M = sum_i m_i J_i J_i^T (lower tri)
        float Mm[6][6];
        #pragma unroll
        for (int j = 0; j < 6; ++j)
            #pragma unroll
            for (int k = 0; k < 6; ++k) Mm[j][k] = 0.0f;

        float J5[6][3];
        #pragma unroll
        for (int i = 0; i < 6; ++i) {
            float Ji[6][3];
            #pragma unroll
            for (int j = 0; j < 6; ++j) {
                if (j <= i) {
                    const float dx = P[i + 1][0] - P[j][0];
                    const float dy = P[i + 1][1] - P[j][1];
                    const float dz = P[i + 1][2] - P[j][2];
                    const float zx = Z[j][0], zy = Z[j][1], zz = Z[j][2];
                    Ji[j][0] = zy * dz - zz * dy;
                    Ji[j][1] = zz * dx - zx * dz;
                    Ji[j][2] = zx * dy - zy * dx;
                }
            }
            const float mi = kMASS[i];
            #pragma unroll
            for (int j = 0; j < 6; ++j) {
                if (j <= i) {
                    #pragma unroll
                    for (int k = 0; k <= j; ++k) {
                        Mm[j][k] += mi * (Ji[j][0] * Ji[k][0] + Ji[j][1] * Ji[k][1] + Ji[j][2] * Ji[k][2]);
                    }
                }
            }
            if (i == 5) {
                #pragma unroll
                for (int j = 0; j < 6; ++j) { J5[j][0] = Ji[j][0]; J5[j][1] = Ji[j][1]; J5[j][2] = Ji[j][2]; }
            }
        }
        #pragma unroll
        for (int j = 0; j < 6; ++j) Mm[j][j] += 0.1f;   // rotor inertia

        // ---- u = (hand - tcp)/||.||,  w = Je^T u  (s = w^T M^-1 w)
        const float dx = hx - P[6][0], dy = hy - P[6][1], dz = hz - P[6][2];
        const float invn = 1.0f / sqrtf(dx * dx + dy * dy + dz * dz);
        const float ux = dx * invn, uy = dy * invn, uz = dz * invn;
        float w[6];
        #pragma unroll
        for (int j = 0; j < 6; ++j) w[j] = J5[j][0] * ux + J5[j][1] * uy + J5[j][2] * uz;

        // ---- Cholesky (M is SPD), store reciprocal diagonals
        float Dinv[6];
        #pragma unroll
        for (int k = 0; k < 6; ++k) {
            float v = Mm[k][k];
            #pragma unroll
            for (int m = 0; m < k; ++m) v -= Mm[k][m] * Mm[k][m];
            const float inv = 1.0f / sqrtf(v);
            Dinv[k] = inv;
            #pragma unroll
            for (int j = k + 1; j < 6; ++j) {
                float t = Mm[j][k];
                #pragma unroll
                for (int m = 0; m < k; ++m) t -= Mm[j][m] * Mm[k][m];
                Mm[j][k] = t * inv;
            }
        }
        // forward/back substitution, single RHS
        float y[6];
        #pragma unroll
        for (int k = 0; k < 6; ++k) {
            float t = w[k];
            #pragma unroll
            for (int m = 0; m < k; ++m) t -= Mm[k][m] * y[m];
            y[k] = t * Dinv[k];
        }
        float xv[6];
        #pragma unroll
        for (int k = 5; k >= 0; --k) {
            float t = y[k];
            #pragma unroll
            for (int m = k + 1; m < 6; ++m) t -= Mm[m][k] * xv[m];
            xv[k] = t * Dinv[k];
        }
        float s = 0.0f;
        #pragma unroll
        for (int j = 0; j < 6; ++j) s += w[j] * xv[j];
        cost = 1.0f / s;
    }

    cbuf[tid] = cost;
    __syncthreads();

    // ---- Per-batch 64-way sum via V_WMMA_F32_16X16X4_F32 with A == ones.
    // D[r,c] = sum_k B[k,c]; reducing d[0] over all 32 lanes counts rows {0,8}
    // => 2 * (sum of all 64 B elements), independent of B's lane layout.
    const int wave = tid >> 5, lane = tid & 31;
    if (wave < 4) {
        v2f b; b.x = cbuf[wave * 64 + lane]; b.y = cbuf[wave * 64 + 32 + lane];
        v2f a; a.x = 1.0f; a.y = 1.0f;
        v8f c = {};
        v8f dacc = __builtin_amdgcn_wmma_f32_16x16x4_f32(
            /*neg_a=*/false, a, /*neg_b=*/false, b,
            /*c_mod=*/(short)0, c, /*reuse_a=*/false, /*reuse_b=*/false);
        float v = dacc[0];
        #pragma unroll
        for (int off = 16; off >= 1; off >>= 1) v += __shfl_xor(v, off, 32);
        if (lane == 0) {
            const int bb = blockIdx.x * 4 + wave;
            if (bb < nBatch) out[bb] = -0.5f * v;
        }
    }
}

extern "C" void kernel_launch(void* const* d_in, const int* in_sizes, int n_in,
                              void* d_out, int out_size, void* d_ws, size_t ws_size,
                              hipStream_t stream) {
    const float* x = (const float*)d_in[0];   // (B, 64, 26) f32; cond/time unused
    float* out = (float*)d_out;               // (B,) f32
    const int nBatch = out_size;              // 2048
    const int totalFloats = in_sizes[0];      // B*64*26
    const int blocks = (nBatch + 3) / 4;      // 4 batches (256 items) per block
    hipLaunchKernelGGL(CostFn_kernel, dim3(blocks), dim3(256), 0, stream,
                       x, out, nBatch, totalFloats);
}